// MPS_67259187855770
// MI455X (gfx1250) — compile-verified
//
#include <hip/hip_runtime.h>
#include <hip/hip_bf16.h>
#include <stdint.h>

typedef float v2f __attribute__((ext_vector_type(2)));
typedef float v8f __attribute__((ext_vector_type(8)));

#define NSITES     64
#define BOND       128
#define BULK_SITES 62
#define SITE_ELEMS (BOND * 2 * BOND)   /* 32768 floats = 128 KB per site */
#define EPSF       1e-30f

/* ---------------- CDNA5 feature probes (compile-safe) ---------------- */

#if __has_builtin(__builtin_amdgcn_wmma_f32_16x16x4_f32)
#define HAVE_WMMA_F32 1
#endif

#if __has_builtin(__builtin_amdgcn_global_load_async_to_lds_b128)
#define HAVE_ASYNC_LDS 1
/* builtin signature (per clang diagnostic): (v4i AS1*, v4i AS3*, imm i32, imm i32) */
typedef int v4i_vs __attribute__((vector_size(16)));
typedef __attribute__((address_space(1))) v4i_vs as1_v4i;
typedef __attribute__((address_space(3))) v4i_vs as3_v4i;
#endif

static __device__ __forceinline__ v8f wmma_f32x4(v2f a, v2f b, v8f c) {
#if defined(HAVE_WMMA_F32)
  /* 8 args: (neg_a, A, neg_b, B, c_mod, C, reuse_a, reuse_b) */
  return __builtin_amdgcn_wmma_f32_16x16x4_f32(false, a, false, b, (short)0, c,
                                               false, false);
#else
  /* placeholder fallback so the TU still compiles; histogram would show wmma=0 */
  #pragma unroll
  for (int j = 0; j < 8; ++j) c[j] += a.x * b.x + a.y * b.y;
  return c;
#endif
}

static __device__ __forceinline__ void async_copy16(const float* g, float* l) {
#if defined(HAVE_ASYNC_LDS)
  __builtin_amdgcn_global_load_async_to_lds_b128(
      (as1_v4i*)(void*)(g), (as3_v4i*)(void*)(l), 0, 0);
#else
  *(float4*)l = *(const float4*)g;
#endif
}

static __device__ __forceinline__ void wait_async_all() {
#if defined(HAVE_ASYNC_LDS)
#if __has_builtin(__builtin_amdgcn_s_wait_asynccnt)
  __builtin_amdgcn_s_wait_asynccnt(0);
#else
  asm volatile("s_wait_asynccnt 0" ::: "memory");
#endif
#endif
}

/* =====================================================================
 * Kernel 2: log-norm (transfer matrix).  One workgroup, 8 waves.
 *   env(128x128) in LDS; per site:
 *     stage1: T[l, v*128+s] = sum_m env[l,m] * A[m*256 + v*128 + s]
 *     stage2: env'[r,s]     = sum_k A[k*128 + r] * T[k*128 + s]   (k = 2l+v)
 *   then max-abs rescale, accumulate log(scale).
 * ===================================================================== */
__global__ __launch_bounds__(256, 1) void mps_lognorm_kernel(
    const float* __restrict__ left, const float* __restrict__ bulk,
    const float* __restrict__ right, float* __restrict__ ws) {
  extern __shared__ float smem[];
  float* env = smem;                    /* 16384 floats */
  float* T   = smem + 16384;            /* 32768 floats */
  float* red = smem + 16384 + 32768;    /* 256 floats   */
  int*   smax = (int*)(red + 256);

  const int tid  = threadIdx.x;         /* 0..255 */
  const int lane = tid & 31;
  const int wave = tid >> 5;            /* 0..7   */
  const int hi   = lane >> 4;           /* 0/1    */
  const int lo   = lane & 15;
  const int r0   = wave * 16;

  float logscale = 0.0f;
  const v8f zf = {0, 0, 0, 0, 0, 0, 0, 0};

  if (tid == 0) *smax = 0;
  __syncthreads();

  /* env0[r,s] = sum_v left[v,r]*left[v,s] */
  float lmax = 0.0f;
  for (int idx = tid; idx < BOND * BOND; idx += 256) {
    const int r = idx >> 7, s = idx & 127;
    const float v = left[r] * left[s] + left[BOND + r] * left[BOND + s];
    env[idx] = v;
    lmax = fmaxf(lmax, fabsf(v));
  }
  atomicMax(smax, __float_as_int(lmax));
  __syncthreads();
  {
    const float scale = fmaxf(__int_as_float(*smax), EPSF);
    const float inv = 1.0f / scale;
    for (int idx = tid; idx < BOND * BOND; idx += 256) env[idx] *= inv;
    logscale += logf(scale);
  }

  for (int i = 0; i < BULK_SITES; ++i) {
    __syncthreads();                    /* env scaled & visible */
    if (tid == 0) *smax = 0;            /* becomes visible at stage1-end barrier */
    const float* A = bulk + (size_t)i * SITE_ELEMS;

    /* ---- stage 1 ---- (M-tile = rows r0..r0+15 of l-dim, N = 256) */
    {
      v8f acc[16];
#pragma unroll
      for (int nt = 0; nt < 16; ++nt) acc[nt] = zf;
      for (int kb = 0; kb < 32; ++kb) {
        const int k0 = 4 * kb + 2 * hi;
        const v2f a = *(const v2f*)(env + (r0 + lo) * BOND + k0);
#pragma unroll
        for (int nt = 0; nt < 16; ++nt) {
          const int col = nt * 16 + lo;
          v2f b;
          b.x = A[k0 * 256 + col];
          b.y = A[(k0 + 1) * 256 + col];
          acc[nt] = wmma_f32x4(a, b, acc[nt]);
        }
      }
#pragma unroll
      for (int nt = 0; nt < 16; ++nt)
#pragma unroll
        for (int j = 0; j < 8; ++j)
          T[(r0 + j + 8 * hi) * 256 + nt * 16 + lo] = acc[nt][j];
    }
    __syncthreads();                    /* T complete, smax reset visible */

    /* ---- stage 2 ---- (M-tile = rows r0..r0+15 of r-dim, K = 256) */
    {
      v8f acc[8];
#pragma unroll
      for (int nt = 0; nt < 8; ++nt) acc[nt] = zf;
      for (int kb = 0; kb < 64; ++kb) {
        const int k0 = 4 * kb + 2 * hi;
        v2f a;                          /* transposed A-operand from global */
        a.x = A[k0 * BOND + r0 + lo];
        a.y = A[(k0 + 1) * BOND + r0 + lo];
#pragma unroll
        for (int nt = 0; nt < 8; ++nt) {
          const int col = nt * 16 + lo;
          v2f b;
          b.x = T[k0 * BOND + col];
          b.y = T[(k0 + 1) * BOND + col];
          acc[nt] = wmma_f32x4(a, b, acc[nt]);
        }
      }
      float lm = 0.0f;
#pragma unroll
      for (int nt = 0; nt < 8; ++nt)
#pragma unroll
        for (int j = 0; j < 8; ++j) lm = fmaxf(lm, fabsf(acc[nt][j]));
      atomicMax(smax, __float_as_int(lm));
      __syncthreads();                  /* all T reads + atomics done */
      const float scale = fmaxf(__int_as_float(*smax), EPSF);
      const float inv = 1.0f / scale;
      logscale += logf(scale);
#pragma unroll
      for (int nt = 0; nt < 8; ++nt)
#pragma unroll
        for (int j = 0; j < 8; ++j)
          env[(r0 + j + 8 * hi) * BOND + nt * 16 + lo] = acc[nt][j] * inv;
    }
  }
  __syncthreads();

  /* final contraction with right: z = sum_{l,v} right[l,v] * (env[l,:].right[:,v]) */
  {
    const int l = tid >> 1, v = tid & 1;
    float t = 0.0f;
    for (int m = 0; m < BOND; ++m) t += env[l * BOND + m] * right[m * 2 + v];
    red[tid] = right[l * 2 + v] * t;
  }
  __syncthreads();
  if (tid == 0) {
    float z = 0.0f;
    for (int t = 0; t < 256; ++t) z += red[t];
    const float scale = fmaxf(fabsf(z), EPSF);
    z /= scale;
    logscale += logf(scale);
    ws[0] = logf(fmaxf(z, EPSF)) + logscale;
  }
}

/* =====================================================================
 * Kernel 1: psi + NLL.  128 blocks x 128 threads (4 waves).
 * Each wave owns a 16-row batch tile (env 16x128 via per-wave LDS buffer).
 * Site matrix (128 KB) double-buffered in LDS via async global->LDS.
 * Per site: P0 = env*A0, P1 = env*A1 by WMMA f32 16x16x4; blend by cfg bit.
 * ===================================================================== */
__global__ __launch_bounds__(128, 1) void mps_psi_kernel(
    const float* __restrict__ left, const float* __restrict__ bulk,
    const float* __restrict__ right, const int* __restrict__ cfg,
    const float* __restrict__ logz_ws, float* __restrict__ out) {
  extern __shared__ float smem[];
  const int tid  = threadIdx.x;
  const int lane = tid & 31;
  const int wave = tid >> 5;            /* 0..3 */
  const int hi   = lane >> 4;
  const int lo   = lane & 15;

  float* envLds = smem + wave * (16 * BOND);          /* 2048 floats/wave */
  float* Abuf0  = smem + 4 * 16 * BOND;               /* 32768 floats      */
  float* Abuf1  = Abuf0 + SITE_ELEMS;                 /* 32768 floats      */

  const int gBase = blockIdx.x * 64 + wave * 16;      /* first batch row   */
  const v8f zf = {0, 0, 0, 0, 0, 0, 0, 0};

  /* per-lane 64-bit config masks for the 8 rows this lane blends (D-layout) */
  unsigned long long mask[8];
#pragma unroll
  for (int j = 0; j < 8; ++j) {
    const int g = gBase + hi * 8 + j;
    const int* crow = cfg + (size_t)g * NSITES;
    unsigned long long m = 0;
    for (int s = 0; s < NSITES; ++s)
      m |= (unsigned long long)(crow[s] & 1) << s;
    mask[j] = m;
  }

  /* env init: env[r,:] = left[cfg[r,0], :]   (per-wave private LDS) */
  for (int idx = lane; idx < 16 * BOND; idx += 32) {
    const int r = idx >> 7, c = idx & 127;
    const int v0 = cfg[(size_t)(gBase + r) * NSITES] & 1;
    envLds[idx] = left[v0 * BOND + c];
  }

  /* prologue: async-stage site 0 into Abuf0 (all 128 threads cooperate) */
  for (int t = 0; t < 64; ++t) {
    const int e4 = tid + t * 128;
    async_copy16(bulk + (size_t)e4 * 4, Abuf0 + (size_t)e4 * 4);
  }

  for (int i = 0; i < BULK_SITES; ++i) {
    float* cur = (i & 1) ? Abuf1 : Abuf0;
    float* nxt = (i & 1) ? Abuf0 : Abuf1;

    wait_async_all();                   /* my async stores to cur complete */
    __syncthreads();                    /* cur full; prev reads of nxt done */

    if (i + 1 < BULK_SITES) {           /* prefetch next site */
      const float* src = bulk + (size_t)(i + 1) * SITE_ELEMS;
      for (int t = 0; t < 64; ++t) {
        const int e4 = tid + t * 128;
        async_copy16(src + (size_t)e4 * 4, nxt + (size_t)e4 * 4);
      }
    }

    const float* A = cur;               /* [k][v][n] : k*256 + v*128 + n */
    v8f acc0[8], acc1[8];
#pragma unroll
    for (int n = 0; n < 8; ++n) { acc0[n] = zf; acc1[n] = zf; }

    for (int kb = 0; kb < 32; ++kb) {
      const int k0 = 4 * kb + 2 * hi;
      const v2f a = *(const v2f*)(envLds + lo * BOND + k0);
#pragma unroll
      for (int n = 0; n < 8; ++n) {
        const int col = n * 16 + lo;
        v2f b0, b1;
        b0.x = A[k0 * 256 + col];
        b0.y = A[(k0 + 1) * 256 + col];
        b1.x = A[k0 * 256 + 128 + col];
        b1.y = A[(k0 + 1) * 256 + 128 + col];
        acc0[n] = wmma_f32x4(a, b0, acc0[n]);
        acc1[n] = wmma_f32x4(a, b1, acc1[n]);
      }
    }

    /* blend by config bit of column (i+1), write env back (D-layout) */
#pragma unroll
    for (int j = 0; j < 8; ++j) {
      const bool sel = ((mask[j] >> (i + 1)) & 1ull) != 0ull;
#pragma unroll
      for (int n = 0; n < 8; ++n) {
        const float val = sel ? acc1[n][j] : acc0[n][j];
        envLds[(j + 8 * hi) * BOND + n * 16 + lo] = val;
      }
    }
  }

  /* psi[r] = env[r,:] . right[:, cfg[r,63]];  out = logZ - log(max(psi^2,eps)) */
  {
    const int r = lo;
    const int g = gBase + r;
    const int c = cfg[(size_t)g * NSITES + (NSITES - 1)] & 1;
    float partial = 0.0f;
    const int kstart = hi * 64;
    for (int k = 0; k < 64; ++k) {
      const int kk = kstart + k;
      partial += envLds[r * BOND + kk] * right[kk * 2 + c];
    }
    const float other = __shfl_xor(partial, 16);
    if (hi == 0) {
      const float psi = partial + other;
      out[g] = logz_ws[0] - logf(fmaxf(psi * psi, EPSF));
    }
  }
}

/* ===================================================================== */
extern "C" void kernel_launch(void* const* d_in, const int* in_sizes, int n_in,
                              void* d_out, int out_size, void* d_ws,
                              size_t ws_size, hipStream_t stream) {
  const float* left  = (const float*)d_in[0];   /* (1,2,128)        */
  const float* bulk  = (const float*)d_in[1];   /* (62,128,2,128)   */
  const float* right = (const float*)d_in[2];   /* (128,2,1)        */
  const int*   cfg   = (const int*)d_in[3];     /* (8192,64)        */
  float* out = (float*)d_out;                   /* (8192,)          */
  float* ws  = (float*)d_ws;                    /* ws[0] = log Z    */

  /* log-norm first (psi kernel consumes ws[0]); same stream => ordered */
  const size_t lds2 = (size_t)(16384 + 32768 + 256) * sizeof(float) + 16;
  mps_lognorm_kernel<<<1, 256, lds2, stream>>>(left, bulk, right, ws);

  const size_t lds1 = (size_t)(4 * 16 * BOND + 2 * SITE_ELEMS) * sizeof(float);
  mps_psi_kernel<<<128, 128, lds1, stream>>>(left, bulk, right, cfg, ws, out);
}